// MambaBlock_70145405878936
// MI455X (gfx1250) — compile-verified
//
#include <hip/hip_runtime.h>

// ---------------------------------------------------------------------------
// Mamba block forward for MI455X (gfx1250), fp32.
// D_MODEL=1024, D_STATE=16, D_CONV=4, EXPAND=2 -> D_INNER=2048, B=4, L=2048
// ---------------------------------------------------------------------------

#define BDIM    4
#define LSEQ    2048
#define DMODEL  1024
#define DINNER  2048
#define DSTATE  16
#define DCONV   4
#define NROWS   (BDIM * LSEQ)          // 8192 "M" rows for the big GEMMs
#define XPW     (1 + 2 * DSTATE)       // 33 columns of the x-projection

typedef __attribute__((ext_vector_type(2))) float v2f;
typedef __attribute__((ext_vector_type(8))) float v8f;
typedef __attribute__((ext_vector_type(4))) unsigned int v4u;
typedef __attribute__((ext_vector_type(8))) int v8i;
typedef __attribute__((ext_vector_type(4))) int v4i;

// ---------------------------------------------------------------------------
// Tensor Data Mover: 2D tile load Global -> LDS with row padding.
//   lds_addr    : byte offset in LDS
//   gaddr       : global address of tile start
//   g1_dw0      : data_size / pad_enable / pad_interval / pad_amount bits
//   tensor_d0/1 : tensor extents (elements) for OOB clamping
//   tile_d0/1   : tile extents (elements); tile_d0 is the contiguous dim
//   stride0     : elements between consecutive rows in global memory
// D# layout per CDNA5 ISA ch.8 (group0: count/type/addresses,
// group1: dims/tile/strides/padding). Groups 2/3 zero (2D tensor).
// ---------------------------------------------------------------------------
__device__ __forceinline__ void tdm_load_2d(
    unsigned int lds_addr, const void* gaddr, unsigned int g1_dw0,
    unsigned int tensor_d0, unsigned int tensor_d1,
    unsigned int tile_d0, unsigned int tile_d1,
    unsigned long long stride0)
{
    const unsigned long long ga = (unsigned long long)gaddr;
    v4u g0;
    g0[0] = 1u;                                    // count=1 (valid), load, user
    g0[1] = lds_addr;                              // LDS byte address
    g0[2] = (unsigned int)ga;                      // global_addr[31:0]
    g0[3] = ((unsigned int)(ga >> 32) & 0x1FFFFFFu) | (2u << 30);  // [56:32] | type=2

    v8i g1;
    g1[0] = (int)g1_dw0;                                           // mask=0|dsize|pad
    g1[1] = (int)(tensor_d0 << 16);                                // dim0[15:0]
    g1[2] = (int)((tensor_d0 >> 16) | (tensor_d1 << 16));          // dim0 hi | dim1 lo
    g1[3] = (int)((tensor_d1 >> 16) | (tile_d0 << 16));            // dim1 hi | tile0
    g1[4] = (int)tile_d1;                                          // tile1 | tile2=0
    g1[5] = (int)(unsigned int)(stride0 & 0xFFFFFFFFull);          // stride0[31:0]
    g1[6] = (int)(unsigned int)(stride0 >> 32);                    // stride0[47:32]
    g1[7] = 0;

    const v4i z4 = {0, 0, 0, 0};
#if __has_include(<hip/amd_detail/amd_gfx1250_TDM.h>)
    const v8i z8 = {0, 0, 0, 0, 0, 0, 0, 0};
    __builtin_amdgcn_tensor_load_to_lds(g0, g1, z4, z4, z8, 0);
#else
    __builtin_amdgcn_tensor_load_to_lds(g0, g1, z4, z4, 0);
#endif
}

// group1 dword0 helpers: data_size=4B (2), pad_enable=1
// A panel: rows of 16 dwords, pad 2 dwords  -> pad_interval=3 (16), pad_amount=1 (2)
#define G1DW0_A (unsigned int)((2u << 16) | (1u << 20) | (3u << 22) | (1u << 25))
// B panel: rows of 128 dwords, pad 4 dwords -> pad_interval=6 (128), pad_amount=3 (4)
#define G1DW0_B (unsigned int)((2u << 16) | (1u << 20) | (6u << 22) | (3u << 25))

#define ASTRIDE 18    // 16 + 2 pad floats
#define BSTRIDE 132   // 128 + 4 pad floats

__device__ __forceinline__ unsigned int lds_off(const void* p) {
    return (unsigned int)(unsigned long long)p;   // LDS aperture: low 32 bits = offset
}

// ---------------------------------------------------------------------------
// fp32 WMMA GEMM: C[M,N] = A[M,K] * B[K,N], row-major, M%128==N%128==0, K%16==0.
// 256 threads = 8 waves; block tile 128x128; wave tile 32x64 (2x4 16x16 frags).
// K panels of 16 DMA'd into double-buffered LDS by the TDM (wave 0 issues,
// s_wait_tensorcnt + block barrier synchronize). A is kept row-major [m][18]
// so a lane's K-pair is one aligned ds_load_b64; B is [k][132] K-major.
// ---------------------------------------------------------------------------
__global__ __launch_bounds__(256) void gemm_f32_wmma_k(
    const float* __restrict__ A, const float* __restrict__ B,
    float* __restrict__ C, int M, int N, int K)
{
    __shared__ __align__(16) float As[2][128 * ASTRIDE];
    __shared__ __align__(16) float Bs[2][16 * BSTRIDE];

    const int tid   = threadIdx.x;
    const int lane  = tid & 31;
    const int wave  = tid >> 5;
    const int hl    = lane >> 4;      // half-wave: 0 or 1
    const int l16   = lane & 15;
    const int waveM = wave & 3;       // 4 waves stacked in M (32 rows each)
    const int waveN = wave >> 2;      // 2 waves in N (64 cols each)
    const long blockM = (long)blockIdx.x * 128;
    const long blockN = (long)blockIdx.y * 128;

    v8f acc[2][4] = {};

    // prologue: DMA first K panel into buffer 0
    if (wave == 0) {
        tdm_load_2d(lds_off(&As[0][0]), &A[blockM * (long)K], G1DW0_A,
                    (unsigned)K, 128u, 16u, 128u, (unsigned long long)K);
        tdm_load_2d(lds_off(&Bs[0][0]), &B[blockN], G1DW0_B,
                    (unsigned)N, 16u, 128u, 16u, (unsigned long long)N);
    }

    int ibuf = 0;
    for (int kb = 0; kb < K; kb += 16) {
        if (wave == 0) __builtin_amdgcn_s_wait_tensorcnt(0);
        __syncthreads();   // panel ready for all waves; other buffer consumed

        // kick off DMA of the next panel into the other buffer
        if (wave == 0 && kb + 16 < K) {
            tdm_load_2d(lds_off(&As[ibuf ^ 1][0]),
                        &A[blockM * (long)K + kb + 16], G1DW0_A,
                        (unsigned)K, 128u, 16u, 128u, (unsigned long long)K);
            tdm_load_2d(lds_off(&Bs[ibuf ^ 1][0]),
                        &B[(long)(kb + 16) * N + blockN], G1DW0_B,
                        (unsigned)N, 16u, 128u, 16u, (unsigned long long)N);
        }

        const float* __restrict__ Ab = As[ibuf];
        const float* __restrict__ Bb = Bs[ibuf];

        // ---- 4 WMMA K-steps of 4 over the 16-wide panel ----
        #pragma unroll
        for (int ks = 0; ks < 4; ++ks) {
            const int krow = ks * 4 + 2 * hl;   // lane's K pair base
            v2f afrag[2];
            v2f bfrag[4];
            #pragma unroll
            for (int i = 0; i < 2; ++i) {
                const int m = waveM * 32 + i * 16 + l16;
                afrag[i] = *(const v2f*)&Ab[m * ASTRIDE + krow];  // ds_load_b64
            }
            #pragma unroll
            for (int j = 0; j < 4; ++j) {
                const int n = waveN * 64 + j * 16 + l16;
                bfrag[j].x = Bb[(krow + 0) * BSTRIDE + n];
                bfrag[j].y = Bb[(krow + 1) * BSTRIDE + n];
            }
            #pragma unroll
            for (int i = 0; i < 2; ++i)
                #pragma unroll
                for (int j = 0; j < 4; ++j)
                    acc[i][j] = __builtin_amdgcn_wmma_f32_16x16x4_f32(
                        false, afrag[i], false, bfrag[j],
                        (short)0, acc[i][j], false, false);
        }
        ibuf ^= 1;
    }

    // ---- store C: D-fragment layout r -> row (r + 8*hl), lane -> col ----
    #pragma unroll
    for (int i = 0; i < 2; ++i)
        #pragma unroll
        for (int j = 0; j < 4; ++j)
            #pragma unroll
            for (int r = 0; r < 8; ++r) {
                const long row = blockM + waveM * 32 + i * 16 + r + 8 * hl;
                const long col = blockN + waveN * 64 + j * 16 + l16;
                C[row * N + col] = acc[i][j][r];
            }
}

// ---------------------------------------------------------------------------
// Causal depthwise conv1d (4 taps) + bias + SiLU.
// ---------------------------------------------------------------------------
__device__ __forceinline__ float silu_f(float v) {
    return v / (1.0f + __expf(-v));
}

__global__ __launch_bounds__(256) void conv_silu_k(
    const float* __restrict__ xz, const float* __restrict__ conv_w,
    const float* __restrict__ conv_b, float* __restrict__ x_conv)
{
    const long idx = (long)blockIdx.x * 256 + threadIdx.x;   // over B*L*DINNER
    const int d  = (int)(idx & (DINNER - 1));
    const long bl = idx >> 11;               // DINNER = 2048 = 2^11
    const int l  = (int)(bl & (LSEQ - 1));
    const long rowbase = bl << 12;           // *4096 into xz

    float acc = conv_b[d];
    #pragma unroll
    for (int j = 0; j < DCONV; ++j) {
        const int ls = l - (DCONV - 1) + j;
        if (ls >= 0) {
            const float xv = xz[rowbase + (long)(ls - l) * 4096 + d];
            acc = fmaf(conv_w[d * DCONV + j], xv, acc);
        }
    }
    x_conv[idx] = silu_f(acc);
}

// ---------------------------------------------------------------------------
// xp[b,l,0:33] = x_conv[b,l,:] @ W_x   (skinny N=33: VALU block reduction)
// ---------------------------------------------------------------------------
__global__ __launch_bounds__(256) void xproj_k(
    const float* __restrict__ x_conv, const float* __restrict__ W_x,
    float* __restrict__ xp)
{
    __shared__ float sm[XPW * 257];
    const int tid = threadIdx.x;
    const long row = blockIdx.x;             // b*L + l

    float acc[XPW];
    #pragma unroll
    for (int n = 0; n < XPW; ++n) acc[n] = 0.0f;

    #pragma unroll
    for (int it = 0; it < DINNER / 256; ++it) {
        const int d = tid + it * 256;
        const float xv = x_conv[row * DINNER + d];
        const float* wrow = &W_x[(long)d * XPW];
        #pragma unroll
        for (int n = 0; n < XPW; ++n)
            acc[n] = fmaf(xv, wrow[n], acc[n]);
    }

    #pragma unroll
    for (int n = 0; n < XPW; ++n) sm[n * 257 + tid] = acc[n];
    __syncthreads();
    for (int s = 128; s > 0; s >>= 1) {
        if (tid < s)
            #pragma unroll
            for (int n = 0; n < XPW; ++n)
                sm[n * 257 + tid] += sm[n * 257 + tid + s];
        __syncthreads();
    }
    if (tid < XPW) xp[row * XPW + tid] = sm[tid * 257];
}

// ---------------------------------------------------------------------------
// Selective scan, fused with softplus(delta), D skip and SiLU(z) gating.
// 4 threads per channel d, each owning 4 of the 16 states; y reduced with
// shfl_xor across the 4-lane group.  Grid: (DINNER/64, B), block 256.
// ---------------------------------------------------------------------------
__device__ __forceinline__ float softplus_f(float v) {
    return (v > 20.0f) ? v : log1pf(__expf(v));
}

__global__ __launch_bounds__(256) void scan_k(
    const float* __restrict__ xp, const float* __restrict__ x_conv,
    const float* __restrict__ xz, const float* __restrict__ W_dt,
    const float* __restrict__ b_dt, const float* __restrict__ A_log,
    const float* __restrict__ D_param, float* __restrict__ y_g)
{
    const int tid = threadIdx.x;
    const int sg  = tid & 3;                  // state group (4 states each)
    const int ch  = tid >> 2;                 // channel within block (0..63)
    const int d   = blockIdx.x * 64 + ch;
    const int b   = blockIdx.y;

    float Arow[4];
    #pragma unroll
    for (int i = 0; i < 4; ++i)
        Arow[i] = -__expf(A_log[(long)d * DSTATE + sg * 4 + i]);

    const float wdt = W_dt[d];
    const float bdt = b_dt[d];
    const float Dp  = D_param[d];

    float h[4] = {0.f, 0.f, 0.f, 0.f};

    for (int l = 0; l < LSEQ; ++l) {
        const long base  = (long)b * LSEQ + l;
        const float* xr  = &xp[base * XPW];
        const float d1   = xr[0];
        const float delta = softplus_f(fmaf(d1, wdt, bdt));
        const float xv   = x_conv[base * DINNER + d];
        const float dx   = delta * xv;

        float y = 0.f;
        #pragma unroll
        for (int i = 0; i < 4; ++i) {
            const float Bv = xr[1 + sg * 4 + i];
            const float Cv = xr[1 + DSTATE + sg * 4 + i];
            const float ab = __expf(delta * Arow[i]);
            h[i] = fmaf(ab, h[i], dx * Bv);
            y = fmaf(h[i], Cv, y);
        }
        y += __shfl_xor(y, 1, 4);
        y += __shfl_xor(y, 2, 4);

        if (sg == 0) {
            const float zv = xz[(base << 12) + DINNER + d];   // z half of xz
            y_g[base * DINNER + d] = (y + xv * Dp) * silu_f(zv);
        }
    }
}

// ---------------------------------------------------------------------------
// Launch
// ---------------------------------------------------------------------------
extern "C" void kernel_launch(void* const* d_in, const int* in_sizes, int n_in,
                              void* d_out, int out_size, void* d_ws, size_t ws_size,
                              hipStream_t stream) {
    const float* x      = (const float*)d_in[0];
    const float* W_in   = (const float*)d_in[1];
    const float* conv_w = (const float*)d_in[2];
    const float* conv_b = (const float*)d_in[3];
    const float* W_x    = (const float*)d_in[4];
    const float* W_dt   = (const float*)d_in[5];
    const float* b_dt   = (const float*)d_in[6];
    const float* A_log  = (const float*)d_in[7];
    const float* D_par  = (const float*)d_in[8];
    const float* W_out  = (const float*)d_in[9];
    float* out = (float*)d_out;

    float* ws     = (float*)d_ws;
    float* xz     = ws;                                    // 8192*4096
    float* x_conv = xz + (long)NROWS * 2 * DINNER;         // 8192*2048
    float* xp     = x_conv + (long)NROWS * DINNER;         // 8192*33
    float* y_g    = xp + (long)NROWS * XPW;                // 8192*2048

    // 1) xz = x @ W_in            [8192,1024]x[1024,4096]
    gemm_f32_wmma_k<<<dim3(NROWS / 128, (2 * DINNER) / 128), 256, 0, stream>>>(
        x, W_in, xz, NROWS, 2 * DINNER, DMODEL);

    // 2) causal depthwise conv + SiLU
    conv_silu_k<<<(NROWS * DINNER) / 256, 256, 0, stream>>>(
        xz, conv_w, conv_b, x_conv);

    // 3) xp = x_conv @ W_x        [8192,2048]x[2048,33]
    xproj_k<<<NROWS, 256, 0, stream>>>(x_conv, W_x, xp);

    // 4) selective scan + gating
    scan_k<<<dim3(DINNER / 64, BDIM), 256, 0, stream>>>(
        xp, x_conv, xz, W_dt, b_dt, A_log, D_par, y_g);

    // 5) out = y_g @ W_out        [8192,2048]x[2048,1024]
    gemm_f32_wmma_k<<<dim3(NROWS / 128, DMODEL / 128), 256, 0, stream>>>(
        y_g, W_out, out, NROWS, DMODEL, DINNER);
}